// OptimizedDynamicSparseAttention_21268678049966
// MI455X (gfx1250) — compile-verified
//
#include <hip/hip_runtime.h>
#include <hip/hip_bf16.h>
#include <stdint.h>

// ---------------- problem constants ----------------
#define CDIM   768
#define C3     2304
#define HEADS  12
#define HD     64
#define SEQ    1024
#define BATCH  4
#define ROWS   (BATCH * SEQ)        // 4096
#define KKEEP  512                  // N * 0.5

typedef float v2f __attribute__((ext_vector_type(2)));
typedef float v8f __attribute__((ext_vector_type(8)));
typedef unsigned v4u __attribute__((ext_vector_type(4)));
typedef int      v8i __attribute__((ext_vector_type(8)));
typedef int      v4i __attribute__((ext_vector_type(4)));

// CDNA5 fp32 WMMA: D(16x16,f32) = A(16x4,f32) x B(4x16,f32) + C
__device__ __forceinline__ v8f wmma4(v2f a, v2f b, v8f c) {
    return __builtin_amdgcn_wmma_f32_16x16x4_f32(
        false, a, false, b, (short)0, c, false, false);
}

// ---- Tensor Data Mover: 2D tile load, global -> LDS ----
// 6-arg builtin on this toolchain:
//   (uint32x4 g0, int32x8 g1, int32x4 g2, int32x4 g3, int32x8 extra, i32 cpol)
// D# layout per CDNA5 ISA.
// Loads nRows x rowLen f32 elements; rows separated by rowStride elements in
// global memory, packed contiguously (rowLen*4 bytes per row) into LDS at lds_off.
__device__ __forceinline__ void tdm_load_2d(unsigned lds_off, const void* gptr,
                                            unsigned rowLen, unsigned nRows,
                                            unsigned rowStride) {
    const unsigned long long ga = (unsigned long long)(uintptr_t)gptr;
    v4u g0;
    g0[0] = 1u;                                           // count=1 (valid user D#)
    g0[1] = lds_off;                                      // LDS byte address
    g0[2] = (unsigned)(ga & 0xFFFFFFFFu);                 // global_addr[31:0]
    g0[3] = (unsigned)((ga >> 32) & 0x01FFFFFFu)          // global_addr[56:32]
            | (2u << 30);                                 // type=2 ("image")
    v8i g1;
    g1[0] = (int)(2u << 16);                              // data_size=2 -> 4 bytes
    g1[1] = (int)((rowLen & 0xFFFFu) << 16);              // tensor_dim0[15:0]
    g1[2] = (int)((rowLen >> 16) | ((nRows & 0xFFFFu) << 16)); // dim0 hi | dim1 lo
    g1[3] = (int)((nRows >> 16) | (rowLen << 16));        // dim1 hi | tile_dim0
    g1[4] = (int)(nRows & 0xFFFFu);                       // tile_dim1 | tile_dim2=0
    g1[5] = (int)rowStride;                               // tensor_dim0_stride lo32
    g1[6] = 0;                                            // stride0 hi | stride1 lo
    g1[7] = 0;                                            // stride1 hi
    const v4i z4 = {0, 0, 0, 0};                          // groups 2/3: dims 2+ unused
    const v8i z8 = {0, 0, 0, 0, 0, 0, 0, 0};
    __builtin_amdgcn_tensor_load_to_lds(g0, g1, z4, z4, z8, 0);
}

// wave32 reductions
__device__ __forceinline__ int wave_sum_i(int v) {
    #pragma unroll
    for (int off = 16; off; off >>= 1) v += __shfl_xor(v, off, 32);
    return v;
}
__device__ __forceinline__ float wave_sum_f(float v) {
    #pragma unroll
    for (int off = 16; off; off >>= 1) v += __shfl_xor(v, off, 32);
    return v;
}
__device__ __forceinline__ float wave_max_f(float v) {
    #pragma unroll
    for (int off = 16; off; off >>= 1) v = fmaxf(v, __shfl_xor(v, off, 32));
    return v;
}

// order-preserving key: larger float -> larger unsigned
__device__ __forceinline__ unsigned okey(float f) {
    unsigned u = __float_as_uint(f);
    return (u & 0x80000000u) ? ~u : (u | 0x80000000u);
}

// ============================================================
// Kernel 1: QKV projection. Y[4096,2304] = X @ Wqkv^T + b, routed to
// Q/K/V [B,H,N,D], Q pre-scaled. A slab (16x768 f32, 48 KB) TDM-staged
// into LDS once per block; each wave computes TWO 16x16 tiles (A reuse).
// ============================================================
__global__ void qkv_gemm_kernel(const float* __restrict__ X,
                                const float* __restrict__ W,
                                const float* __restrict__ bias,
                                const float* __restrict__ temp,
                                float* __restrict__ Qb,
                                float* __restrict__ Kb,
                                float* __restrict__ Vb) {
    __shared__ float sA[16 * CDIM];                 // 48 KB
    const int lane  = threadIdx.x & 31;
    const int wave  = threadIdx.x >> 5;
    const int mtile = blockIdx.y;                   // 0..255
    const int nt0   = (blockIdx.x * 8 + wave) * 2;  // 0..142 step 2
    const int mn    = lane & 15;
    const int kb    = (lane >> 4) * 2;

    if (wave == 0) {
        tdm_load_2d(0u, X + (size_t)mtile * 16 * CDIM, CDIM, 16, CDIM);
        __builtin_amdgcn_s_wait_tensorcnt(0);
    }
    __syncthreads();

    const float* aL    = sA + mn * CDIM;
    const float* brow0 = W + (size_t)(nt0 * 16 + mn) * CDIM;
    const float* brow1 = W + (size_t)((nt0 + 1) * 16 + mn) * CDIM;

    v8f acc0 = {}, acc1 = {};
    #pragma unroll 4
    for (int k = 0; k < CDIM; k += 4) {
        __builtin_prefetch(brow0 + k + 64, 0, 0);
        v2f a  = *(const v2f*)(aL + k + kb);
        v2f b0 = *(const v2f*)(brow0 + k + kb);
        v2f b1 = *(const v2f*)(brow1 + k + kb);
        acc0 = wmma4(a, b0, acc0);
        acc1 = wmma4(a, b1, acc1);
    }

    float t = temp[0];
    t = (t < 0.01f) ? 0.01f : t;
    const float qs = 0.125f / t;                    // HD^-0.5 / clip(temp)

    #pragma unroll
    for (int r = 0; r < 8; ++r) {
        const int mm   = (lane < 16) ? r : (r + 8);
        const int orow = mtile * 16 + mm;
        const int b_   = orow >> 10;
        const int n_   = orow & 1023;
        #pragma unroll
        for (int half = 0; half < 2; ++half) {
            const int ocol = (nt0 + half) * 16 + mn;
            const float v  = ((half == 0) ? acc0[r] : acc1[r]) + bias[ocol];
            const int sec    = ocol / CDIM;
            const int within = ocol - sec * CDIM;
            const int h      = within >> 6;
            const int d      = within & 63;
            const size_t idx = (((size_t)(b_ * HEADS + h)) * SEQ + n_) * HD + d;
            if (sec == 0)      Qb[idx] = v * qs;
            else if (sec == 1) Kb[idx] = v;
            else               Vb[idx] = v;
        }
    }
}

// ============================================================
// Kernel 2: per (b*h, 16-query tile) attention.
//   TDM-stage Q tile; S = Q @ K^T via WMMA into 64 KB LDS slab;
//   exact 512th-largest per row (radix select) + masked softmax;
//   P @ V via WMMA with V TDM-pipelined in 64x64 chunks per K-half.
// LDS map (floats): sS[16384] | pP[2048] | sQ[1024] | bV0[4096] | bV1[4096]
// ============================================================
#define OFF_P_F  16384
#define OFF_Q_F  18432
#define OFF_V0_F 19456
#define OFF_V1_F 23552
#define SMEM_F   27648          // 110,592 bytes (< 320 KB/WGP)

__global__ void attn_kernel(const float* __restrict__ Qb,
                            const float* __restrict__ Kb,
                            const float* __restrict__ Vb,
                            float* __restrict__ attn) {
    extern __shared__ float smem[];
    float* sS = smem;
    float* pP = smem + OFF_P_F;
    float* sQ = smem + OFF_Q_F;

    const int lane = threadIdx.x & 31;
    const int wave = threadIdx.x >> 5;
    const int bh   = blockIdx.y;       // 0..47
    const int qt   = blockIdx.x;       // 0..63
    const int b_   = bh / HEADS;
    const int h    = bh % HEADS;

    const float* Qh = Qb + (size_t)bh * SEQ * HD;
    const float* Kh = Kb + (size_t)bh * SEQ * HD;
    const float* Vh = Vb + (size_t)bh * SEQ * HD;

    const int mn = lane & 15;
    const int kb = (lane >> 4) * 2;

    // ---- TDM-stage the Q tile (16x64 f32 = 4 KB) ----
    if (wave == 0) {
        tdm_load_2d(OFF_Q_F * 4u, Qh + (size_t)(qt * 16) * HD, HD, 16, HD);
        __builtin_amdgcn_s_wait_tensorcnt(0);
    }
    __syncthreads();

    // hoist the 16 Q A-fragments; reused across all 8 column blocks
    v2f aq[16];
    #pragma unroll
    for (int t = 0; t < 16; ++t)
        aq[t] = *(const v2f*)(sQ + mn * HD + t * 4 + kb);

    // ---- scores: S = Qtile @ K^T ----
    for (int cb = wave; cb < 64; cb += 8) {
        const float* krow = Kh + (size_t)(cb * 16 + mn) * HD;
        v8f acc = {};
        #pragma unroll
        for (int t = 0; t < 16; ++t) {
            v2f b = *(const v2f*)(krow + t * 4 + kb);
            acc = wmma4(aq[t], b, acc);
        }
        #pragma unroll
        for (int r = 0; r < 8; ++r) {
            const int mm = (lane < 16) ? r : (r + 8);
            sS[mm * 1024 + cb * 16 + mn] = acc[r];
        }
    }
    __syncthreads();

    // ---- per-row exact kth-largest + softmax (wave handles 2 rows) ----
    for (int rr = 0; rr < 2; ++rr) {
        const int rowi = wave + rr * 8;
        float    vals[32];
        unsigned keys[32];
        #pragma unroll
        for (int i = 0; i < 32; ++i) {
            const float f = sS[rowi * 1024 + lane + i * 32];
            vals[i] = f;
            keys[i] = okey(f);
        }
        // bit-descent: largest T with |{key >= T}| >= KKEEP == 512th largest key
        unsigned T = 0u;
        for (int bit = 31; bit >= 0; --bit) {
            const unsigned trial = T | (1u << bit);
            int c = 0;
            #pragma unroll
            for (int i = 0; i < 32; ++i) c += (keys[i] >= trial) ? 1 : 0;
            c = wave_sum_i(c);
            if (c >= KKEEP) T = trial;
        }
        float mx = -3.402823466e38f;
        #pragma unroll
        for (int i = 0; i < 32; ++i) mx = fmaxf(mx, vals[i]);
        mx = wave_max_f(mx);

        float e[32];
        float s = 0.f;
        #pragma unroll
        for (int i = 0; i < 32; ++i) {
            const float ev = (keys[i] >= T) ? __expf(vals[i] - mx) : 0.f;
            e[i] = ev;
            s += ev;
        }
        s = wave_sum_f(s);
        const float inv = 1.f / s;
        #pragma unroll
        for (int i = 0; i < 32; ++i)
            sS[rowi * 1024 + lane + i * 32] = e[i] * inv;
    }
    __syncthreads();

    // ---- PV: Out16x64 = P[16,1024] @ V[1024,64] ----
    // waves split (16-col tile of D) x (K-half); V streamed via TDM in
    // 64x64 f32 (16 KB) chunks, one staging buffer per K-half.
    {
        const int dt = wave & 3;
        const int kh = wave >> 2;
        const int d0 = dt * 16;
        const float* vbuf = smem + (kh ? OFF_V1_F : OFF_V0_F);  // scalar select (no LDS ptr array)
        v8f acc = {};
        for (int c = 0; c < 8; ++c) {
            __syncthreads();   // previous chunk fully consumed before overwrite
            if (wave == 0) {
                tdm_load_2d(OFF_V0_F * 4u, Vh + (size_t)(c * 64) * HD, HD, 64, HD);
                __builtin_amdgcn_s_wait_tensorcnt(0);
            } else if (wave == 4) {
                tdm_load_2d(OFF_V1_F * 4u, Vh + (size_t)(512 + c * 64) * HD, HD, 64, HD);
                __builtin_amdgcn_s_wait_tensorcnt(0);
            }
            __syncthreads();
            const float* prow = sS + mn * 1024 + kh * 512 + c * 64;
            #pragma unroll
            for (int kk = 0; kk < 64; kk += 4) {
                v2f a = *(const v2f*)(prow + kk + kb);
                v2f b;
                b.x = vbuf[(kk + kb)     * HD + d0 + mn];
                b.y = vbuf[(kk + kb + 1) * HD + d0 + mn];
                acc = wmma4(a, b, acc);
            }
        }
        #pragma unroll
        for (int r = 0; r < 8; ++r) {
            const int mm = (lane < 16) ? r : (r + 8);
            pP[kh * 1024 + mm * 64 + d0 + mn] = acc[r];
        }
    }
    __syncthreads();

    // ---- combine K-halves, scatter to attn[B,N,C] ----
    for (int idx = threadIdx.x; idx < 1024; idx += 256) {
        const int mm = idx >> 6;
        const int d  = idx & 63;
        const float v = pP[idx] + pP[1024 + idx];
        attn[((size_t)(b_ * SEQ) + qt * 16 + mm) * CDIM + h * HD + d] = v;
    }
}

// ============================================================
// Kernel 3: output projection. out[4096,768] = attn @ Wp^T + bp
// Same TDM-staged-A, 2-way N-blocked WMMA GEMM.
// ============================================================
__global__ void proj_gemm_kernel(const float* __restrict__ A,
                                 const float* __restrict__ W,
                                 const float* __restrict__ bias,
                                 float* __restrict__ out) {
    __shared__ float sA[16 * CDIM];                 // 48 KB
    const int lane  = threadIdx.x & 31;
    const int wave  = threadIdx.x >> 5;
    const int mtile = blockIdx.y;                   // 0..255
    const int nt0   = (blockIdx.x * 8 + wave) * 2;  // 0..46 step 2
    const int mn    = lane & 15;
    const int kb    = (lane >> 4) * 2;

    if (wave == 0) {
        tdm_load_2d(0u, A + (size_t)mtile * 16 * CDIM, CDIM, 16, CDIM);
        __builtin_amdgcn_s_wait_tensorcnt(0);
    }
    __syncthreads();

    const float* aL    = sA + mn * CDIM;
    const float* brow0 = W + (size_t)(nt0 * 16 + mn) * CDIM;
    const float* brow1 = W + (size_t)((nt0 + 1) * 16 + mn) * CDIM;

    v8f acc0 = {}, acc1 = {};
    #pragma unroll 4
    for (int k = 0; k < CDIM; k += 4) {
        __builtin_prefetch(brow0 + k + 64, 0, 0);
        v2f a  = *(const v2f*)(aL + k + kb);
        v2f b0 = *(const v2f*)(brow0 + k + kb);
        v2f b1 = *(const v2f*)(brow1 + k + kb);
        acc0 = wmma4(a, b0, acc0);
        acc1 = wmma4(a, b1, acc1);
    }

    #pragma unroll
    for (int r = 0; r < 8; ++r) {
        const int mm   = (lane < 16) ? r : (r + 8);
        const int orow = mtile * 16 + mm;
        out[(size_t)orow * CDIM + nt0 * 16 + mn]        = acc0[r] + bias[nt0 * 16 + mn];
        out[(size_t)orow * CDIM + (nt0 + 1) * 16 + mn]  = acc1[r] + bias[(nt0 + 1) * 16 + mn];
    }
}

// ============================================================
extern "C" void kernel_launch(void* const* d_in, const int* in_sizes, int n_in,
                              void* d_out, int out_size, void* d_ws, size_t ws_size,
                              hipStream_t stream) {
    (void)in_sizes; (void)n_in; (void)out_size; (void)ws_size;

    const float* x      = (const float*)d_in[0];
    const float* qkv_w  = (const float*)d_in[1];
    const float* qkv_b  = (const float*)d_in[2];
    const float* proj_w = (const float*)d_in[3];
    const float* proj_b = (const float*)d_in[4];
    const float* temp   = (const float*)d_in[5];
    float*       out    = (float*)d_out;

    const size_t perBuf = (size_t)BATCH * HEADS * SEQ * HD;   // 3,145,728 floats
    float* Qb   = (float*)d_ws;
    float* Kb   = Qb + perBuf;
    float* Vb   = Kb + perBuf;
    float* attn = Vb + perBuf;                                // [B,N,C]

    dim3 blk(256);

    // 1) QKV projection + head split + q scaling  (144 ntiles -> 72 pairs / 8 waves)
    qkv_gemm_kernel<<<dim3(9, ROWS / 16), blk, 0, stream>>>(
        x, qkv_w, qkv_b, temp, Qb, Kb, Vb);

    // 2) top-k sparse attention (108 KB dynamic LDS; CDNA5 WGP allows 320 KB)
    attn_kernel<<<dim3(SEQ / 16, BATCH * HEADS), blk, SMEM_F * sizeof(float), stream>>>(
        Qb, Kb, Vb, attn);

    // 3) output projection  (48 ntiles -> 24 pairs / 8 waves)
    proj_gemm_kernel<<<dim3(3, ROWS / 16), blk, 0, stream>>>(
        attn, proj_w, proj_b, out);
}